// Linear_15891378995769
// MI455X (gfx1250) — compile-verified
//
#include <hip/hip_runtime.h>

typedef __attribute__((ext_vector_type(2))) float v2f;
typedef __attribute__((ext_vector_type(4))) float v4f;
typedef __attribute__((ext_vector_type(8))) float v8f;

#define SIZE   64
#define BATCH  8192
#define TSPAN  512
#define TOUT   (TSPAN + 1)          // 513
#define CLEN   32                   // time-steps per chunk
#define CHUNKS (TSPAN / CLEN)       // 16
#define NTILES (BATCH / 16)         // 512 batch tiles of 16 rows
#define P2_WAVES 8
#define P2_BLOCK (32 * P2_WAVES)

static_assert(NTILES == 512, "chunk extraction assumes NTILES==512");
static_assert(CHUNKS * CLEN == TSPAN, "chunking must tile the time span");

// ---------------------------------------------------------------------------
// Phase 1: one workgroup computes A_c = (W^T)^(c*CLEN), c = 0..CHUNKS-1,
// into workspace. Serial 64x64x64 products in f32 VALU (tiny: ~12 MFLOP,
// 0.03% of total work). 1024 threads: thread (m = tid>>4, n0 = tid&15)
// computes 4 outputs per matrix product.
// ---------------------------------------------------------------------------
__global__ __launch_bounds__(1024) void koopman_chunk_powers(
    const float* __restrict__ W, float* __restrict__ Apow) {
  __shared__ float WT[SIZE * SIZE];
  __shared__ float BUF0[SIZE * SIZE];
  __shared__ float BUF1[SIZE * SIZE];
  __shared__ float BUF2[SIZE * SIZE];
  const int tid = threadIdx.x;
  const int m  = tid >> 4;
  const int n0 = tid & 15;

  // WT[k][n] = W[n][k]
  for (int i = tid; i < SIZE * SIZE; i += 1024)
    WT[i] = W[(i & 63) * SIZE + (i >> 6)];
  // A_0 = I
  for (int i = tid; i < SIZE * SIZE; i += 1024)
    Apow[i] = ((i >> 6) == (i & 63)) ? 1.0f : 0.0f;
  __syncthreads();
  for (int i = tid; i < SIZE * SIZE; i += 1024) BUF0[i] = WT[i];
  __syncthreads();

  // P = WT^CLEN via CLEN-1 serial products, ping-pong BUF0/BUF1
  float* cur = BUF0;
  float* nxt = BUF1;
  for (int it = 0; it < CLEN - 1; ++it) {
    float a0 = 0.f, a1 = 0.f, a2 = 0.f, a3 = 0.f;
    for (int k = 0; k < SIZE; ++k) {
      const float a = cur[m * SIZE + k];
      a0 += a * WT[k * SIZE + n0     ];
      a1 += a * WT[k * SIZE + n0 + 16];
      a2 += a * WT[k * SIZE + n0 + 32];
      a3 += a * WT[k * SIZE + n0 + 48];
    }
    nxt[m * SIZE + n0     ] = a0;
    nxt[m * SIZE + n0 + 16] = a1;
    nxt[m * SIZE + n0 + 32] = a2;
    nxt[m * SIZE + n0 + 48] = a3;
    __syncthreads();
    float* t = cur; cur = nxt; nxt = t;
  }

  // A_1 = P
  for (int i = tid; i < SIZE * SIZE; i += 1024)
    Apow[SIZE * SIZE + i] = cur[i];

  // A_c = A_{c-1} @ P, c = 2..CHUNKS-1 (P preserved; ping-pong nxt/BUF2)
  float* P    = cur;
  float* ping = nxt;
  float* pong = BUF2;
  float* prev = P;
  for (int c = 2; c < CHUNKS; ++c) {
    float* dst = (c & 1) ? pong : ping;
    float a0 = 0.f, a1 = 0.f, a2 = 0.f, a3 = 0.f;
    for (int k = 0; k < SIZE; ++k) {
      const float a = prev[m * SIZE + k];
      a0 += a * P[k * SIZE + n0     ];
      a1 += a * P[k * SIZE + n0 + 16];
      a2 += a * P[k * SIZE + n0 + 32];
      a3 += a * P[k * SIZE + n0 + 48];
    }
    dst[m * SIZE + n0     ] = a0;
    dst[m * SIZE + n0 + 16] = a1;
    dst[m * SIZE + n0 + 32] = a2;
    dst[m * SIZE + n0 + 48] = a3;
    const size_t base = (size_t)c * SIZE * SIZE + (size_t)m * SIZE;
    Apow[base + n0     ] = a0;
    Apow[base + n0 + 16] = a1;
    Apow[base + n0 + 32] = a2;
    Apow[base + n0 + 48] = a3;
    __syncthreads();
    prev = dst;
  }
}

// ---------------------------------------------------------------------------
// Phase 2: wave w = (tile, chunk) computes S_c = x0_tile @ A_c, then rolls
// x <- x @ W^T for 31 (last chunk: 32) steps; every state is written.
// Per step: 64x v_wmma_f32_16x16x4_f32 (4 n-tiles x 16 k-chunks), a
// wave-private LDS round trip for the D-layout -> A-layout shuffle, and the
// same LDS tile read back as float4 rows for fully-coalesced b128 stores
// (output stores are the roofline bound: 1.07 GB @ 23.3 TB/s).
// B operands are read from LDS at each use (ds_load_2addr-friendly, LDS
// broadcast across waves) instead of pinning 128 VGPRs: keeps the live set
// under 256 VGPRs so occupancy hides the per-step WMMA/LDS latency chain.
// Wave32 operand layouts per CDNA5 ISA 7.12.2:
//   A(16x4 f32): lane = M (&15), K-half split at lane 16, v2f = {K, K+1}
//   B(4x16 f32): lane = N (&15), K-half split at lane 16, v2f = {K, K+1}
//   C/D(16x16):  VGPR r: lanes 0-15 -> M=r, lanes 16-31 -> M=r+8, N = lane&15
// ---------------------------------------------------------------------------
__global__ __launch_bounds__(P2_BLOCK) void koopman_rollout(
    const float* __restrict__ x0, const float* __restrict__ W,
    const float* __restrict__ Apow, float* __restrict__ out) {
  __shared__ __align__(16) float WT[SIZE * SIZE];
  __shared__ __align__(16) float XB[P2_WAVES][16 * SIZE];

  const int tid  = threadIdx.x;
  const int lane = tid & 31;
  const int wv   = tid >> 5;

  for (int i = tid; i < SIZE * SIZE; i += P2_BLOCK)
    WT[i] = W[(i & 63) * SIZE + (i >> 6)];          // WT[k][n] = W[n][k]
  __syncthreads();

  const int gw    = blockIdx.x * P2_WAVES + wv;     // 0..8191
  const int tile  = gw & (NTILES - 1);
  const int chunk = gw >> 9;                        // blocks never straddle chunks
  const int r0    = tile * 16;

  const int half  = lane >> 4;
  const int l15   = lane & 15;
  const int khalf = half * 2;                       // K offset for this lane half
  const int mbase = half * 8;                       // D-layout M offset

  const v8f vzero = {0.f, 0.f, 0.f, 0.f, 0.f, 0.f, 0.f, 0.f};
  v8f acc[4];
  float* xb = &XB[wv][0];

  // ---- S_c = x0_tile @ A_chunk ----
  {
    v2f a[16];
#pragma unroll
    for (int c = 0; c < 16; ++c)
      a[c] = *(const v2f*)(x0 + (size_t)(r0 + l15) * SIZE + 4 * c + khalf);
    const float* Ac = Apow + (size_t)chunk * SIZE * SIZE;
#pragma unroll
    for (int n = 0; n < 4; ++n) acc[n] = vzero;
#pragma unroll
    for (int c = 0; c < 16; ++c) {
      const int kb = 4 * c + khalf;
#pragma unroll
      for (int n = 0; n < 4; ++n) {
        v2f b;
        b.x = Ac[(kb    ) * SIZE + 16 * n + l15];
        b.y = Ac[(kb + 1) * SIZE + 16 * n + l15];
        acc[n] = __builtin_amdgcn_wmma_f32_16x16x4_f32(
            false, a[c], false, b, (short)0, acc[n], false, false);
      }
    }
  }

  // ---- stage S_c tile in LDS, then coalesced b128 store at t = tbase ----
  const int tbase = chunk * CLEN;
#pragma unroll
  for (int n = 0; n < 4; ++n)
#pragma unroll
    for (int r = 0; r < 8; ++r)
      xb[(mbase + r) * SIZE + 16 * n + l15] = acc[n][r];

  asm volatile("s_wait_dscnt 0" ::: "memory");  // within-wave LDS RAW hand-off

#pragma unroll
  for (int p = 0; p < 8; ++p) {
    const int row = 2 * p + half;               // rows 0..15 across lane halves
    const v4f sv = *(const v4f*)(xb + row * SIZE + l15 * 4);
    *(v4f*)(out + ((size_t)(r0 + row) * TOUT + tbase) * SIZE + l15 * 4) = sv;
  }

  // ---- rollout: each chunk owns t = tbase+1 .. tbase+nsteps (no overlap) ----
  const int nsteps = (chunk == CHUNKS - 1) ? CLEN : (CLEN - 1);
#pragma unroll 1
  for (int s = 1; s <= nsteps; ++s) {
    // A operands for this step from the tile staged in LDS
    v2f a[16];
#pragma unroll
    for (int c = 0; c < 16; ++c)
      a[c] = *(const v2f*)(xb + l15 * SIZE + 4 * c + khalf);

    asm volatile("" ::: "memory");  // keep the xb overwrite below after these loads

#pragma unroll
    for (int n = 0; n < 4; ++n) acc[n] = vzero;
#pragma unroll
    for (int c = 0; c < 16; ++c) {
      const int kb = 4 * c + khalf;
#pragma unroll
      for (int n = 0; n < 4; ++n) {
        v2f b;  // W^T B-operand straight from LDS (pair 64 floats apart)
        b.x = WT[(kb    ) * SIZE + 16 * n + l15];
        b.y = WT[(kb + 1) * SIZE + 16 * n + l15];
        acc[n] = __builtin_amdgcn_wmma_f32_16x16x4_f32(
            false, a[c], false, b, (short)0, acc[n], false, false);
      }
    }

    // D layout -> row-major tile in wave-private LDS (feeds next step's A
    // operands AND the coalesced store below)
#pragma unroll
    for (int n = 0; n < 4; ++n)
#pragma unroll
      for (int r = 0; r < 8; ++r)
        xb[(mbase + r) * SIZE + 16 * n + l15] = acc[n][r];

    asm volatile("s_wait_dscnt 0" ::: "memory");  // LDS in-order per wave

    // coalesced output: each b128 store covers two contiguous 256B rows
    const int t = tbase + s;
#pragma unroll
    for (int p = 0; p < 8; ++p) {
      const int row = 2 * p + half;
      const v4f sv = *(const v4f*)(xb + row * SIZE + l15 * 4);
      *(v4f*)(out + ((size_t)(r0 + row) * TOUT + t) * SIZE + l15 * 4) = sv;
    }
  }
}

extern "C" void kernel_launch(void* const* d_in, const int* in_sizes, int n_in,
                              void* d_out, int out_size, void* d_ws, size_t ws_size,
                              hipStream_t stream) {
  (void)in_sizes; (void)n_in; (void)out_size; (void)ws_size;
  const float* x0 = (const float*)d_in[0];   // [8192, 64] f32
  const float* W  = (const float*)d_in[1];   // [64, 64] f32
  float* out  = (float*)d_out;               // [8192, 513, 64] f32
  float* Apow = (float*)d_ws;                // CHUNKS * 64*64 f32 = 256 KB scratch

  koopman_chunk_powers<<<1, 1024, 0, stream>>>(W, Apow);
  koopman_rollout<<<(NTILES * CHUNKS) / P2_WAVES, P2_BLOCK, 0, stream>>>(
      x0, W, Apow, out);
}